// UoPAttention_1614907703446
// MI455X (gfx1250) — compile-verified
//
#include <hip/hip_runtime.h>
#include <math.h>
#include <stdint.h>

typedef __attribute__((ext_vector_type(2))) float v2f;
typedef __attribute__((ext_vector_type(8))) float v8f;
typedef __attribute__((ext_vector_type(4))) unsigned int v4u;
typedef __attribute__((ext_vector_type(4))) int v4i;
typedef __attribute__((ext_vector_type(8))) int v8i;

#define B_  64
#define S_  2048
#define D1_ 256
#define D2_ 256
#define P_  256
#define SCHUNK 128            // s-rows per workgroup in the GEMM kernel
#define NCHUNKS (S_ / SCHUNK) // 16

#if __has_builtin(__builtin_amdgcn_tensor_load_to_lds) && \
    __has_builtin(__builtin_amdgcn_s_wait_tensorcnt)
#define USE_TDM 1
#else
#define USE_TDM 0
#endif

// ---------------------------------------------------------------------------
// Kernel 1: attr[b][p] = input_1[b,:] @ W_attr[:,p] + b_attr[p]   (tiny GEMM)
// ---------------------------------------------------------------------------
__global__ void attr_kernel(const float* __restrict__ in1,
                            const float* __restrict__ Wa,
                            const float* __restrict__ ba,
                            float* __restrict__ attr) {
  const int b = blockIdx.x;
  const int p = threadIdx.x;
  float acc = ba[p];
  const float* row = in1 + b * D1_;
  #pragma unroll 8
  for (int d = 0; d < D1_; ++d)
    acc = fmaf(row[d], Wa[d * P_ + p], acc);
  attr[b * P_ + p] = acc;
}

// ---------------------------------------------------------------------------
// Kernel 2 (WMMA + TDM + split-softmax): for one (b, 128-row s-chunk):
//   w_s   = tanh(attr[b] + in2_tile @ W_text + b_text) . w_pool + b_pool
//   m_c   = max_s w_s  (masked -> -1e9)
//   z_c   = sum_s e^{w_s - m_c}
//   v_c[d]= sum_s e^{w_s - m_c} * in2_tile[s][d]   <- tile reused from LDS,
//                                                     so input_2 is read ONCE.
// 512 threads = 16 waves. Wave w owns N-tile (cols 16w..16w+15), 8 M-tiles.
// Input tile (128x256 f32 = 128 KB) staged into LDS by the Tensor Data Mover.
// ---------------------------------------------------------------------------
__global__ void weights_kernel(const float* __restrict__ in2,
                               const int*   __restrict__ mask,
                               const float* __restrict__ Wt,
                               const float* __restrict__ bt,
                               const float* __restrict__ attr,
                               const float* __restrict__ wpool,
                               const float* __restrict__ bpool,
                               float* __restrict__ mbuf,   // [B*NCHUNKS]
                               float* __restrict__ zbuf,   // [B*NCHUNKS]
                               float* __restrict__ vbuf) { // [B*NCHUNKS*D2]
  extern __shared__ float smem[];
  float* tile = smem;                        // SCHUNK * D2_ floats (128 KB)
  float* wacc = smem + SCHUNK * D2_;         // SCHUNK floats: raw weights
  float* wexp = wacc + SCHUNK;               // SCHUNK floats: e^{w - m_c}

  const int b     = blockIdx.x >> 4;         // NCHUNKS == 16
  const int chunk = blockIdx.x & 15;
  const int sbase = chunk * SCHUNK;
  const int tid   = threadIdx.x;
  const int wave  = tid >> 5;
  const int lane  = tid & 31;

  const float* gsrc = in2 + ((size_t)b * S_ + sbase) * D2_;

#if USE_TDM
  // ---- Tensor Data Mover: DMA the whole 128x256 f32 tile into LDS ----
  if (tid == 0) {
    const uint64_t ga = (uint64_t)(uintptr_t)gsrc;
    // Flat LDS aperture: addr[31:0] is the LDS byte offset
    const uint32_t lds_base = (uint32_t)(uintptr_t)tile;
    v4u g0;
    g0[0] = 1u;                                   // count=1, user descriptor
    g0[1] = lds_base;                             // lds_addr
    g0[2] = (uint32_t)ga;                         // global_addr[31:0]
    g0[3] = (uint32_t)((ga >> 32) & 0x01FFFFFFu)  // global_addr[56:32]
            | (2u << 30);                         // type = 2 ("image")
    v8i g1;
    g1[0] = (int)(2u << 16);                      // data_size = 4 bytes
    g1[1] = (int)((uint32_t)D2_ << 16);           // tensor_dim0 = 256 (lo16<<16)
    g1[2] = (int)((uint32_t)SCHUNK << 16);        // dim0 hi=0 | tensor_dim1=128
    g1[3] = (int)((uint32_t)D2_ << 16);           // dim1 hi=0 | tile_dim0 = 256
    g1[4] = SCHUNK;                               // tile_dim1=128, tile_dim2=0
    g1[5] = D2_;                                  // tensor_dim0_stride = 256
    g1[6] = 0;                                    // stride hi | dim1_stride lo
    g1[7] = 0;                                    // tensor_dim1_stride hi
    v4i z = {};                                   // groups 2/3 unused (2-D tile)
#if defined(__clang_major__) && (__clang_major__ >= 23)
    v8i z8 = {};
    __builtin_amdgcn_tensor_load_to_lds(g0, g1, z, z, z8, 0);
#else
    __builtin_amdgcn_tensor_load_to_lds(g0, g1, z, z, 0);
#endif
  }
  if (tid < SCHUNK) wacc[tid] = 0.0f;
  __builtin_amdgcn_s_wait_tensorcnt(0);   // issuing wave's TENSORcnt -> 0
  __syncthreads();                        // publish LDS tile to all waves
#else
  // ---- fallback: classic float4 staging ----
  #pragma unroll
  for (int i = 0; i < (SCHUNK * D2_) / (512 * 4); ++i) {   // 16 iters
    const int f = tid + i * 512;
    ((float4*)tile)[f] = ((const float4*)gsrc)[f];
  }
  if (tid < SCHUNK) wacc[tid] = 0.0f;
  __syncthreads();
#endif

  const int ncol0 = wave * 16;         // this wave's N-tile base column
  const int halfk = lane >> 4;         // 0: K0/K1 half, 1: K2/K3 half
  const int l15   = lane & 15;

  v8f acc[8] = {};                     // 8 M-tiles of 16x16 f32 C

  for (int k0 = 0; k0 < D2_; k0 += 4) {
    const int kk = k0 + 2 * halfk;
    // B fragment: 4x16 slab of W_text (rows kk,kk+1 for this lane half)
    v2f bfrag;
    bfrag.x = Wt[kk * P_ + ncol0 + l15];
    bfrag.y = Wt[(kk + 1) * P_ + ncol0 + l15];
    // Batch A-fragment LDS loads, then issue the 8 WMMAs.
    v2f afr[8];
    #pragma unroll
    for (int m = 0; m < 8; ++m) {
      const float2 av = *(const float2*)&tile[(m * 16 + l15) * D2_ + kk];
      afr[m].x = av.x; afr[m].y = av.y;
    }
    #pragma unroll
    for (int m = 0; m < 8; ++m)
      acc[m] = __builtin_amdgcn_wmma_f32_16x16x4_f32(
          false, afr[m], false, bfrag, (short)0, acc[m], false, false);
  }

  // ---- fused epilogue on C layout: VGPR r, lane L -> M=r+8*(L>=16), N=L&15
  const int p   = ncol0 + l15;
  const float add = bt[p] + attr[b * P_ + p];
  const float wpp = wpool[p];
  #pragma unroll
  for (int m = 0; m < 8; ++m) {
    #pragma unroll
    for (int r = 0; r < 8; ++r) {
      float h = tanhf(acc[m][r] + add);
      float contrib = h * wpp;
      // reduce over the 16 N-lanes of this half (xor masks stay in-half)
      contrib += __shfl_xor(contrib, 1, 32);
      contrib += __shfl_xor(contrib, 2, 32);
      contrib += __shfl_xor(contrib, 4, 32);
      contrib += __shfl_xor(contrib, 8, 32);
      if (l15 == 0)
        atomicAdd(&wacc[m * 16 + r + 8 * halfk], contrib); // ds_add_f32
    }
  }
  __syncthreads();

  // ---- mask + bias, chunk-local softmax stats ----
  if (tid < SCHUNK) {
    float val = wacc[tid] + bpool[0];
    if (mask[b * S_ + sbase + tid] == 0) val = -1e9f;
    wacc[tid] = val;
  }
  __syncthreads();

  // chunk max (uniform LDS-broadcast loop, trivial vs the GEMM)
  float mc = -1e30f;
  #pragma unroll 8
  for (int r = 0; r < SCHUNK; ++r) mc = fmaxf(mc, wacc[r]);
  if (tid < SCHUNK) wexp[tid] = __expf(wacc[tid] - mc);
  __syncthreads();

  float zc = 0.0f;
  #pragma unroll 8
  for (int r = 0; r < SCHUNK; ++r) zc += wexp[r];

  // partial weighted vector v_c[d] = sum_r wexp[r] * tile[r][d]
  // (tile reused from LDS -- avoids re-reading input_2 from HBM)
  if (tid < D2_) {
    float accv = 0.0f;
    #pragma unroll 4
    for (int r = 0; r < SCHUNK; ++r)
      accv = fmaf(wexp[r], tile[r * D2_ + tid], accv);
    vbuf[(size_t)(b * NCHUNKS + chunk) * D2_ + tid] = accv;
  }
  if (tid == 0) {
    mbuf[b * NCHUNKS + chunk] = mc;
    zbuf[b * NCHUNKS + chunk] = zc;
  }
}

// ---------------------------------------------------------------------------
// Kernel 3: combine split-softmax partials (tiny: reads ~1 MB, not 128 MB)
//   M = max_c m_c ; Z = sum_c z_c e^{m_c-M} ; out = sum_c v_c e^{m_c-M} / Z
// ---------------------------------------------------------------------------
__global__ void finalize_kernel(const float* __restrict__ mbuf,
                                const float* __restrict__ zbuf,
                                const float* __restrict__ vbuf,
                                float* __restrict__ out) {
  const int b = blockIdx.x;
  const int d = threadIdx.x;
  float M = -1e30f;
  #pragma unroll
  for (int c = 0; c < NCHUNKS; ++c) M = fmaxf(M, mbuf[b * NCHUNKS + c]);
  float Z = 0.0f, accv = 0.0f;
  #pragma unroll
  for (int c = 0; c < NCHUNKS; ++c) {
    const float sc = __expf(mbuf[b * NCHUNKS + c] - M);
    Z    = fmaf(zbuf[b * NCHUNKS + c], sc, Z);
    accv = fmaf(vbuf[(size_t)(b * NCHUNKS + c) * D2_ + d], sc, accv);
  }
  out[b * D2_ + d] = accv / Z;
}

// ---------------------------------------------------------------------------
extern "C" void kernel_launch(void* const* d_in, const int* in_sizes, int n_in,
                              void* d_out, int out_size, void* d_ws, size_t ws_size,
                              hipStream_t stream) {
  (void)in_sizes; (void)n_in; (void)out_size; (void)ws_size;
  const float* in1  = (const float*)d_in[0];
  const float* in2  = (const float*)d_in[1];
  const int*   mask = (const int*)  d_in[2];
  const float* Wa   = (const float*)d_in[3];
  const float* ba   = (const float*)d_in[4];
  const float* Wt   = (const float*)d_in[5];
  const float* bt   = (const float*)d_in[6];
  const float* wp   = (const float*)d_in[7];
  const float* bp   = (const float*)d_in[8];
  float* out  = (float*)d_out;

  float* attr = (float*)d_ws;                       // 64*256
  float* mbuf = attr + B_ * P_;                     // 64*16
  float* zbuf = mbuf + B_ * NCHUNKS;                // 64*16
  float* vbuf = zbuf + B_ * NCHUNKS;                // 64*16*256 (1 MB)

  attr_kernel<<<B_, P_, 0, stream>>>(in1, Wa, ba, attr);

  const size_t smem2 = (size_t)(SCHUNK * D2_ + 2 * SCHUNK) * sizeof(float); // ~129 KB
  weights_kernel<<<B_ * NCHUNKS, 512, smem2, stream>>>(in2, mask, Wt, bt,
                                                       attr, wp, bp,
                                                       mbuf, zbuf, vbuf);

  finalize_kernel<<<B_, D2_, 0, stream>>>(mbuf, zbuf, vbuf, out);
}